// DfMap_15109694947345
// MI455X (gfx1250) — compile-verified
//
#include <hip/hip_runtime.h>
#include <hip/hip_bf16.h>

typedef __attribute__((ext_vector_type(2))) float v2f;
typedef __attribute__((ext_vector_type(8))) float v8f;

#define DD 64
#define HW (64 * 64)
#define DHW (64 * 64 * 64)      // 262144 = 2^18
#define NCH 12                  // 3*IN_CH output channels
#define NMAPS 8                 // BS * IN_CH * 3 / 3
#define NSTEPS 6
#define EPSBN 1e-5f
#define INVSCALE 0.015625f      // 1 / 2^6
#define WRAWN (NCH * 4 * 27)    // 1296 raw conv weights

__device__ __forceinline__ int iclamp63(int v) {
    return v < 0 ? 0 : (v > 63 ? 63 : v);
}

// ---------------------------------------------------------------------------
// Kernel 0: zero the BN reduction buffer (ws is poisoned once, never reset)
// ---------------------------------------------------------------------------
__global__ void k_zero_stats(float* __restrict__ stats) {
    if (threadIdx.x < 24) stats[threadIdx.x] = 0.f;
}

// ---------------------------------------------------------------------------
// Kernel 1: implicit-GEMM conv3d via V_WMMA_F32_16X16X4_F32
//   tile: M = 16 consecutive x positions, N = 16 (12 real out-ch), K = 4 in-ch
//   27 chained WMMAs (one per 3x3x3 tap), bias preloaded into C.
//   Weights staged LDS-side through the gfx1250 async-to-LDS DMA path.
//   BN stats: LDS float atomics (ds_add_f32) -> one global atomic per block.
// ---------------------------------------------------------------------------
__global__ __launch_bounds__(256) void k_conv_wmma(
    const float* __restrict__ f,      // (2, 4, 64,64,64)
    const float* __restrict__ w,      // (12, 4, 3,3,3)
    const float* __restrict__ bias,   // (12)
    float* __restrict__ convout,      // (2, 12, 64,64,64)
    float* __restrict__ gsum,         // (12)
    float* __restrict__ gsumsq)       // (12)
{
    __shared__ float wraw[WRAWN];         // raw (12,4,27) weights
    __shared__ float wlds[27 * 4 * 16];   // [tap][in_ch][out_ch16], padded
    __shared__ float ssum[12];
    __shared__ float ssumsq[12];

    const int tid = threadIdx.x;

    // Stage 1: async DMA the raw weight block into LDS (ASYNCcnt path).
    for (int i = tid; i < WRAWN; i += 256) {
        unsigned lds_off = (unsigned)(uintptr_t)&wraw[i];
        unsigned long long gaddr = (unsigned long long)(uintptr_t)(w + i);
        asm volatile("global_load_async_to_lds_b32 %0, %1, off"
                     :: "v"(lds_off), "v"(gaddr) : "memory");
    }
    asm volatile("s_wait_asynccnt 0x0" ::: "memory");
    __syncthreads();

    // Stage 2: pad/transpose into the WMMA-friendly [tap][c][o16] layout.
    for (int i = tid; i < 27 * 64; i += 256) {
        const int tap = i >> 6;
        const int rem = i & 63;
        const int c = rem >> 4;
        const int o = rem & 15;
        wlds[i] = (o < NCH) ? wraw[(o * 4 + c) * 27 + tap] : 0.f;
    }
    if (tid < 12) { ssum[tid] = 0.f; ssumsq[tid] = 0.f; }
    __syncthreads();

    // One wave = one 16x16 output tile. 8 waves per block.
    const int wave = (blockIdx.x << 3) + (tid >> 5);
    const int lane = tid & 31;
    const int m    = lane & 15;     // spatial row within tile (A-matrix M)
    const int half = lane >> 4;     // lane half selects K pair / M group
    const int o    = lane & 15;     // output channel column (B/C/D N)

    const int x0 = (wave & 3) << 4;
    const int y  = (wave >> 2) & 63;
    const int z  = (wave >> 8) & 63;
    const int n  = wave >> 14;

    v8f acc;
    const float bv = (o < NCH) ? bias[o] : 0.f;
#pragma unroll
    for (int v = 0; v < 8; ++v) acc[v] = bv;

    const float* fn = f + (size_t)n * 4 * DHW;
    const int c0 = half << 1;       // K = 2*half + reg  (ISA 16x4 f32 A layout)

#pragma unroll
    for (int tap = 0; tap < 27; ++tap) {
        const int kz = tap / 9;
        const int ky = (tap / 3) % 3;
        const int kx = tap % 3;
        const int zz = z + kz - 1;
        const int yy = y + ky - 1;
        const int xx = x0 + m + kx - 1;
        const bool ok = ((unsigned)zz < 64u) & ((unsigned)yy < 64u) & ((unsigned)xx < 64u);
        const int sp = ((iclamp63(zz) << 6) | iclamp63(yy)) << 6 | iclamp63(xx);

        v2f a, bfrag;
        a[0] = ok ? fn[(c0 + 0) * DHW + sp] : 0.f;
        a[1] = ok ? fn[(c0 + 1) * DHW + sp] : 0.f;
        bfrag[0] = wlds[tap * 64 + (c0 + 0) * 16 + o];
        bfrag[1] = wlds[tap * 64 + (c0 + 1) * 16 + o];

        acc = __builtin_amdgcn_wmma_f32_16x16x4_f32(
            /*neg_a=*/false, a, /*neg_b=*/false, bfrag,
            /*c_mod=*/(short)0, acc, /*reuse_a=*/false, /*reuse_b=*/false);
    }

    if (o < NCH) {
        // C/D layout: VGPR v holds M = v + 8*half  ->  x = x0 + 8*half + v
        float* outp = convout +
            ((((size_t)n * NCH + o) * 64 + z) * 64 + y) * 64 + x0 + (half << 3);
        float4 lo = make_float4(acc[0], acc[1], acc[2], acc[3]);
        float4 hi = make_float4(acc[4], acc[5], acc[6], acc[7]);
        *(float4*)(outp + 0) = lo;
        *(float4*)(outp + 4) = hi;

        float s1 = 0.f, s2 = 0.f;
#pragma unroll
        for (int v = 0; v < 8; ++v) { s1 += acc[v]; s2 += acc[v] * acc[v]; }
        atomicAdd(&ssum[o], s1);        // ds_add_f32
        atomicAdd(&ssumsq[o], s2);
    }
    __syncthreads();
    if (tid < 12) {
        unsafeAtomicAdd(&gsum[tid], ssum[tid]);
        unsafeAtomicAdd(&gsumsq[tid], ssumsq[tid]);
    }
}

// ---------------------------------------------------------------------------
// Kernel 2: fold batchnorm + 1/2^NSTEPS into per-channel scale/shift
// ---------------------------------------------------------------------------
__global__ void k_finalize_stats(
    const float* __restrict__ gsum, const float* __restrict__ gsumsq,
    const float* __restrict__ gamma, const float* __restrict__ beta,
    float* __restrict__ scale, float* __restrict__ shift)
{
    const int c = threadIdx.x;
    if (c < NCH) {
        const float invcnt = 1.f / (float)(2 * DHW);
        const float mean = gsum[c] * invcnt;
        const float var  = gsumsq[c] * invcnt - mean * mean;
        const float inv  = rsqrtf(var + EPSBN);
        const float g    = gamma[c] * inv;
        scale[c] = g * INVSCALE;
        shift[c] = (beta[c] - mean * g) * INVSCALE;
    }
}

// ---------------------------------------------------------------------------
// Kernel 3: vec0 = convout * scale[ch] + shift[ch]   (float4 vectorized)
// ---------------------------------------------------------------------------
__global__ __launch_bounds__(256) void k_normalize(
    const float* __restrict__ convout,
    const float* __restrict__ scale, const float* __restrict__ shift,
    float* __restrict__ vec)
{
    const int i = blockIdx.x * 256 + threadIdx.x;   // float4 index
    const int ch = (i >> 16) % NCH;                 // (4*i)>>18 = i>>16
    const float s = scale[ch], t = shift[ch];
    float4 v = ((const float4*)convout)[i];
    v.x = fmaf(v.x, s, t);
    v.y = fmaf(v.y, s, t);
    v.z = fmaf(v.z, s, t);
    v.w = fmaf(v.w, s, t);
    ((float4*)vec)[i] = v;
}

// ---------------------------------------------------------------------------
// 8-corner trilinear gather helpers (border: zero weight, clamped index)
// ---------------------------------------------------------------------------
__device__ __forceinline__ void trilerp3(
    const float* __restrict__ c0, const float* __restrict__ c1,
    const float* __restrict__ c2,
    float cz, float cy, float cx, float& o0, float& o1, float& o2)
{
    const float z0 = floorf(cz), y0 = floorf(cy), x0 = floorf(cx);
    const float fz = cz - z0, fy = cy - y0, fx = cx - x0;
    const int iz = (int)z0, iy = (int)y0, ix = (int)x0;
    o0 = 0.f; o1 = 0.f; o2 = 0.f;
#pragma unroll
    for (int k = 0; k < 8; ++k) {
        const int dz = k >> 2, dy = (k >> 1) & 1, dx = k & 1;
        const int zi = iz + dz, yi = iy + dy, xi = ix + dx;
        const float wgt = (dz ? fz : 1.f - fz) * (dy ? fy : 1.f - fy) * (dx ? fx : 1.f - fx);
        const bool ok = ((unsigned)zi < 64u) & ((unsigned)yi < 64u) & ((unsigned)xi < 64u);
        const int idx = ((iclamp63(zi) << 6) | iclamp63(yi)) << 6 | iclamp63(xi);
        const float wv = ok ? wgt : 0.f;
        o0 = fmaf(wv, c0[idx], o0);
        o1 = fmaf(wv, c1[idx], o1);
        o2 = fmaf(wv, c2[idx], o2);
    }
}

__device__ __forceinline__ float trilerp1(
    const float* __restrict__ src, float cz, float cy, float cx)
{
    const float z0 = floorf(cz), y0 = floorf(cy), x0 = floorf(cx);
    const float fz = cz - z0, fy = cy - y0, fx = cx - x0;
    const int iz = (int)z0, iy = (int)y0, ix = (int)x0;
    float o = 0.f;
#pragma unroll
    for (int k = 0; k < 8; ++k) {
        const int dz = k >> 2, dy = (k >> 1) & 1, dx = k & 1;
        const int zi = iz + dz, yi = iy + dy, xi = ix + dx;
        const float wgt = (dz ? fz : 1.f - fz) * (dy ? fy : 1.f - fy) * (dx ? fx : 1.f - fx);
        const bool ok = ((unsigned)zi < 64u) & ((unsigned)yi < 64u) & ((unsigned)xi < 64u);
        const int idx = ((iclamp63(zi) << 6) | iclamp63(yi)) << 6 | iclamp63(xi);
        o = fmaf(ok ? wgt : 0.f, src[idx], o);
    }
    return o;
}

// ---------------------------------------------------------------------------
// Kernel 4: one scaling-and-squaring step fused with the final resample.
//   newvec(p) = vec(p) + trilerp(vec, grid(p) + vec(p))
//   outmap(p) = trilerp1(f_q, grid(p) + newvec(p))
// ---------------------------------------------------------------------------
__global__ __launch_bounds__(256) void k_step_and_map(
    const float* __restrict__ vin,    // (8, 3, 64^3)
    float* __restrict__ vout,         // (8, 3, 64^3)
    const float* __restrict__ f,      // (8, 64^3) view of (2,4,64^3)
    float* __restrict__ outmap)       // (8, 64^3) slab of d_out for this step
{
    const int tid = blockIdx.x * 256 + threadIdx.x;
    const int q = tid >> 18;                 // which deformation map (0..7)
    const int p = tid & (DHW - 1);
    const int z = p >> 12, y = (p >> 6) & 63, x = p & 63;

    const float* c0 = vin + (size_t)q * 3 * DHW;
    const float* c1 = c0 + DHW;
    const float* c2 = c1 + DHW;
    const float* src = f + (size_t)q * DHW;

    // Displacements are small; the identity cacheline covers most corners.
    __builtin_prefetch(src + p, 0, 0);       // global_prefetch_b8

    const float v0 = c0[p], v1 = c1[p], v2 = c2[p];

    float s0, s1, s2;
    trilerp3(c0, c1, c2, (float)z + v0, (float)y + v1, (float)x + v2, s0, s1, s2);

    const float n0 = v0 + s0, n1 = v1 + s1, n2 = v2 + s2;
    float* d0 = vout + (size_t)q * 3 * DHW;
    d0[p] = n0;
    d0[DHW + p] = n1;
    d0[2 * DHW + p] = n2;

    outmap[(size_t)q * DHW + p] =
        trilerp1(src, (float)z + n0, (float)y + n1, (float)x + n2);
}

// ---------------------------------------------------------------------------
extern "C" void kernel_launch(void* const* d_in, const int* in_sizes, int n_in,
                              void* d_out, int out_size, void* d_ws, size_t ws_size,
                              hipStream_t stream) {
    const float* f      = (const float*)d_in[0];   // (2,4,64,64,64)
    const float* conv_w = (const float*)d_in[1];   // (12,4,3,3,3)
    const float* conv_b = (const float*)d_in[2];   // (12)
    const float* gamma  = (const float*)d_in[3];   // (12)
    const float* beta   = (const float*)d_in[4];   // (12)
    float* out = (float*)d_out;                    // (6,2,4,64,64,64)
    float* ws  = (float*)d_ws;

    const size_t VECN = (size_t)2 * NCH * DHW;     // 6,291,456 floats
    float* convout = ws;                           // 25.2 MB
    float* vecA    = ws + VECN;                    // 25.2 MB
    float* vecB    = ws + 2 * VECN;                // 25.2 MB
    float* stats   = ws + 3 * VECN;                // sum[12] sumsq[12] scale[12] shift[12]

    k_zero_stats<<<1, 64, 0, stream>>>(stats);

    // 32768 tiles (2*64*64*4), 8 waves/block -> 4096 blocks
    k_conv_wmma<<<4096, 256, 0, stream>>>(f, conv_w, conv_b, convout,
                                          stats, stats + 12);

    k_finalize_stats<<<1, 32, 0, stream>>>(stats, stats + 12, gamma, beta,
                                           stats + 24, stats + 36);

    k_normalize<<<(int)(VECN / 4 / 256), 256, 0, stream>>>(convout, stats + 24,
                                                           stats + 36, vecA);

    float* vin = vecA;
    float* vout = vecB;
    for (int s = 0; s < NSTEPS; ++s) {
        k_step_and_map<<<(NMAPS * DHW) / 256, 256, 0, stream>>>(
            vin, vout, f, out + (size_t)s * NMAPS * DHW);
        float* t = vin; vin = vout; vout = t;
    }
}